// TetraSampler_78881369358966
// MI455X (gfx1250) — compile-verified
//
#include <hip/hip_runtime.h>

// ---------------------------------------------------------------------------
// TetraSampler on MI455X (gfx1250, wave32).
// Phase 1 (intersection) reformulated as fp32 WMMA GEMM:
//   per-ray features r = [o(3), d(3), m=o x d(3), 1, 0, 0]   (K = 12)
//   per-tri coeffs (4 fields): a, u_num, v_num, t_num are all r . c_tri
// Uses V_WMMA_F32_16X16X4_F32 chained 3x (K=12) for each of the 4 fields.
// Phase 2 (tet walk) is scalar per-ray (gather bound).
// Divisions replaced by v_rcp_f32 (results only feed compares / min-reduce).
// ---------------------------------------------------------------------------

#define N_RAYS   4096
#define NTRI     12288
#define NTET     3072
#define MAXS     128
#define NTILE    (NTRI / 16)          // 768 triangle tiles
#define STEPSZ   (0.05f / 128.0f)
#define HIT_T    5.0f

typedef __attribute__((ext_vector_type(2))) float v2f;
typedef __attribute__((ext_vector_type(8))) float v8f;

struct F3 { float x, y, z; };
__device__ __forceinline__ F3 mkf3(float a, float b, float c) { F3 r; r.x=a; r.y=b; r.z=c; return r; }
__device__ __forceinline__ F3 sub3(F3 a, F3 b) { return mkf3(a.x-b.x, a.y-b.y, a.z-b.z); }
__device__ __forceinline__ F3 cross3(F3 a, F3 b) {
    return mkf3(a.y*b.z - a.z*b.y, a.z*b.x - a.x*b.z, a.x*b.y - a.y*b.x);
}
__device__ __forceinline__ float dot3(F3 a, F3 b) { return a.x*b.x + a.y*b.y + a.z*b.z; }
__device__ __forceinline__ F3 ld3(const float* p, int i) { return mkf3(p[3*i+0], p[3*i+1], p[3*i+2]); }
__device__ __forceinline__ float fastrcp(float x) { return __builtin_amdgcn_rcpf(x); }  // v_rcp_f32

// ---------------------------------------------------------------------------
// Kernel 0: build per-triangle WMMA B-matrix coefficients.
// Layout: Bws[(((tile*4 + field)*3 + chunk)*32 + lane)*2 + vr]
//   K index of (chunk, vr, lane): k = chunk*4 + vr + 2*(lane>=16)
// One tile = 16 triangles = 768 floats (3 KB).
// ---------------------------------------------------------------------------
__global__ __launch_bounds__(256) void ts_tri_setup(const float* __restrict__ verts,
                                                    const int*   __restrict__ faces,
                                                    float*       __restrict__ Bws) {
    int tri = blockIdx.x * blockDim.x + threadIdx.x;
    if (tri >= NTRI) return;
    int i0 = faces[tri*3+0], i1 = faces[tri*3+1], i2 = faces[tri*3+2];
    F3 p0 = ld3(verts, i0), p1 = ld3(verts, i1), p2 = ld3(verts, i2);
    F3 e1 = sub3(p1, p0), e2 = sub3(p2, p0);
    F3 n  = cross3(e1, e2);
    F3 e2xp0 = cross3(e2, p0);
    F3 p0xe1 = cross3(p0, e1);

    float cf[4][12];
#pragma unroll
    for (int F = 0; F < 4; F++)
#pragma unroll
        for (int k = 0; k < 12; k++) cf[F][k] = 0.0f;

    // field 0: a      = -d.n
    cf[0][3] = -n.x;      cf[0][4] = -n.y;      cf[0][5] = -n.z;
    // field 1: u_num  = -(e2 x p0).d + e2.m
    cf[1][3] = -e2xp0.x;  cf[1][4] = -e2xp0.y;  cf[1][5] = -e2xp0.z;
    cf[1][6] =  e2.x;     cf[1][7] =  e2.y;     cf[1][8] =  e2.z;
    // field 2: v_num  = -(p0 x e1).d - e1.m
    cf[2][3] = -p0xe1.x;  cf[2][4] = -p0xe1.y;  cf[2][5] = -p0xe1.z;
    cf[2][6] = -e1.x;     cf[2][7] = -e1.y;     cf[2][8] = -e1.z;
    // field 3: t_num  = n.o - (p0.n)*1
    cf[3][0] =  n.x;      cf[3][1] =  n.y;      cf[3][2] =  n.z;
    cf[3][9] = -dot3(p0, n);

    int tile = tri >> 4;
    int col  = tri & 15;
#pragma unroll
    for (int F = 0; F < 4; F++) {
#pragma unroll
        for (int k = 0; k < 12; k++) {
            int c = k >> 2, r = k & 3;
            int vr = r & 1, hf = r >> 1;
            int lane = col + 16 * hf;
            Bws[(size_t)(((tile*4 + F)*3 + c)*32 + lane)*2 + vr] = cf[F][k];
        }
    }
}

// ---------------------------------------------------------------------------
// Kernel 1: WMMA intersection. One wave handles a 16-ray tile against all
// 768 triangle tiles. 12 v_wmma_f32_16x16x4_f32 per triangle tile
// (4 fields x 3 chained K-chunks). D layout: lane holds column N=lane%16,
// rows 8*(lane/16)..+7 -> elementwise postprocess + shfl_xor min-reduction.
// ---------------------------------------------------------------------------
__global__ __launch_bounds__(256) void ts_intersect(const float* __restrict__ ro,
                                                    const float* __restrict__ rd,
                                                    const float* __restrict__ Bws,
                                                    float*       __restrict__ tminWS,
                                                    int*         __restrict__ fidxWS) {
    int gtid = blockIdx.x * blockDim.x + threadIdx.x;
    int wave = gtid >> 5;            // 0..255 : one 16-ray tile per wave
    int lane = gtid & 31;
    int col  = lane & 15;
    int half = lane >> 4;            // 0 or 1
    int rbase = wave * 16;

    // per-lane ray features for ray M = col
    F3 o = ld3(ro, rbase + col);
    F3 d = ld3(rd, rbase + col);
    F3 m = cross3(o, d);

    v2f fp0, fp1, fp2, fp3, fp4, fp5;
    fp0.x = o.x; fp0.y = o.y;
    fp1.x = o.z; fp1.y = d.x;
    fp2.x = d.y; fp2.y = d.z;
    fp3.x = m.x; fp3.y = m.y;
    fp4.x = m.z; fp4.y = 1.0f;
    fp5.x = 0.0f; fp5.y = 0.0f;
    // A-matrix 16x4 f32 layout: lane<16 -> K={4c,4c+1}; lane>=16 -> K={4c+2,4c+3}
    v2f A0 = half ? fp1 : fp0;
    v2f A1 = half ? fp3 : fp2;
    v2f A2 = half ? fp5 : fp4;

    float tmin[8];
    int   tidx[8];
#pragma unroll
    for (int v = 0; v < 8; v++) { tmin[v] = 1.0e10f; tidx[v] = 0; }

    for (int tile = 0; tile < NTILE; ++tile) {
        const float* bt = Bws + (size_t)tile * 768;
        __builtin_prefetch(bt + 768, 0, 0);            // global_prefetch_b8 next tile
        const v2f* bv = (const v2f*)bt;

        v8f accA = {}, accU = {}, accV = {}, accT = {};
        // field a
        accA = __builtin_amdgcn_wmma_f32_16x16x4_f32(false, A0, false, bv[(0*3+0)*32 + lane], (short)0, accA, false, false);
        accA = __builtin_amdgcn_wmma_f32_16x16x4_f32(false, A1, false, bv[(0*3+1)*32 + lane], (short)0, accA, false, false);
        accA = __builtin_amdgcn_wmma_f32_16x16x4_f32(false, A2, false, bv[(0*3+2)*32 + lane], (short)0, accA, false, false);
        // field u_num
        accU = __builtin_amdgcn_wmma_f32_16x16x4_f32(false, A0, false, bv[(1*3+0)*32 + lane], (short)0, accU, false, false);
        accU = __builtin_amdgcn_wmma_f32_16x16x4_f32(false, A1, false, bv[(1*3+1)*32 + lane], (short)0, accU, false, false);
        accU = __builtin_amdgcn_wmma_f32_16x16x4_f32(false, A2, false, bv[(1*3+2)*32 + lane], (short)0, accU, false, false);
        // field v_num
        accV = __builtin_amdgcn_wmma_f32_16x16x4_f32(false, A0, false, bv[(2*3+0)*32 + lane], (short)0, accV, false, false);
        accV = __builtin_amdgcn_wmma_f32_16x16x4_f32(false, A1, false, bv[(2*3+1)*32 + lane], (short)0, accV, false, false);
        accV = __builtin_amdgcn_wmma_f32_16x16x4_f32(false, A2, false, bv[(2*3+2)*32 + lane], (short)0, accV, false, false);
        // field t_num
        accT = __builtin_amdgcn_wmma_f32_16x16x4_f32(false, A0, false, bv[(3*3+0)*32 + lane], (short)0, accT, false, false);
        accT = __builtin_amdgcn_wmma_f32_16x16x4_f32(false, A1, false, bv[(3*3+1)*32 + lane], (short)0, accT, false, false);
        accT = __builtin_amdgcn_wmma_f32_16x16x4_f32(false, A2, false, bv[(3*3+2)*32 + lane], (short)0, accT, false, false);

        int triN = (tile << 4) + col;
#pragma unroll
        for (int v = 0; v < 8; v++) {
            float a  = accA[v], un = accU[v], vn = accV[v], tn = accT[v];
            bool  anz = fabsf(a) > 1e-9f;
            float aa  = anz ? a : 1e-9f;
            float f   = fastrcp(aa);                   // v_rcp_f32 (feeds compares only)
            float u   = f * un;
            float vv  = f * vn;
            float t   = f * tn;
            bool ok = anz && (u >= 0.0f) && (u <= 1.0f) && (vv >= 0.0f) &&
                      (u + vv <= 1.0f) && (t > 1e-6f);
            if (ok && (t < tmin[v])) { tmin[v] = t; tidx[v] = triN; }
        }
    }

    // min+argmin reduction across the 16 lanes of each half (wave32 shuffles)
#pragma unroll
    for (int s = 8; s >= 1; s >>= 1) {
#pragma unroll
        for (int v = 0; v < 8; v++) {
            float ot = __shfl_xor(tmin[v], s, 32);
            int   oi = __shfl_xor(tidx[v], s, 32);
            if (ot < tmin[v] || (ot == tmin[v] && oi < tidx[v])) { tmin[v] = ot; tidx[v] = oi; }
        }
    }
    if (col == 0) {
#pragma unroll
        for (int v = 0; v < 8; v++) {
            int ray = rbase + 8 * half + v;
            tminWS[ray] = tmin[v];
            fidxWS[ray] = tidx[v];
        }
    }
}

// ---------------------------------------------------------------------------
// Kernel 2: tet walk + sampling. One thread per ray.
// ---------------------------------------------------------------------------
__device__ __forceinline__ void bary4(const float* __restrict__ verts,
                                      const int*   __restrict__ tetras,
                                      F3 p, int tet, float w[4]) {
    const int4* tt = (const int4*)tetras;
    int4 id = tt[tet];
    F3 v0 = ld3(verts, id.x);
    F3 v1 = ld3(verts, id.y);
    F3 v2 = ld3(verts, id.z);
    F3 v3 = ld3(verts, id.w);
    F3 c1 = sub3(v1, v0), c2 = sub3(v2, v0), c3 = sub3(v3, v0);
    F3 b  = sub3(p, v0);
    F3 c2xc3 = cross3(c2, c3);
    float det = dot3(c1, c2xc3);
    float inv = fastrcp(det);                          // v_rcp_f32
    float w1 = dot3(b, c2xc3) * inv;
    float w2 = dot3(c1, cross3(b, c3)) * inv;
    float w3 = dot3(c1, cross3(c2, b)) * inv;
    w[0] = 1.0f - w1 - w2 - w3; w[1] = w1; w[2] = w2; w[3] = w3;
}

__global__ __launch_bounds__(256) void ts_sample(const float* __restrict__ verts,
                                                 const float* __restrict__ ro,
                                                 const float* __restrict__ rd,
                                                 const int*   __restrict__ tetras,
                                                 const int*   __restrict__ topo,
                                                 const float* __restrict__ tminWS,
                                                 const int*   __restrict__ fidxWS,
                                                 float* __restrict__ out) {
    int ray = blockIdx.x * blockDim.x + threadIdx.x;
    if (ray >= N_RAYS) return;

    const size_t RS = (size_t)N_RAYS * MAXS;
    float* o_ray = out;
    float* o_tet = out + RS;
    float* o_bar = out + 2 * RS;
    float* o_ts  = out + 6 * RS;
    float* o_te  = out + 6 * RS + N_RAYS;
    float* o_pos = out + 6 * RS + 2 * N_RAYS;

    F3 o = ld3(ro, ray);
    F3 d = ld3(rd, ray);
    float tm   = tminWS[ray];
    int   fidx = fidxWS[ray];
    bool  hit  = tm < HIT_T;
    float t0   = hit ? tm : 0.0f;
    F3 hp = mkf3(o.x + t0 * d.x, o.y + t0 * d.y, o.z + t0 * d.z);
    int  tet   = hit ? (fidx >> 2) : 0;
    bool alive = hit;

    o_ts[ray] = t0;
    o_te[ray] = hit ? (t0 + 0.05f) : 0.0f;

    for (int k = 0; k < MAXS; ++k) {
        float tk = (float)k * STEPSZ + 1.0e-4f;
        F3 p = mkf3(hp.x + tk * d.x, hp.y + tk * d.y, hp.z + tk * d.z);

        float w[4];
#pragma unroll
        for (int it = 0; it < 3; ++it) {
            bary4(verts, tetras, p, tet, w);
            bool inside = (w[0] >= -1e-6f) && (w[1] >= -1e-6f) && (w[2] >= -1e-6f) && (w[3] >= -1e-6f);
            int am = 0; float mv = w[0];
            if (w[1] < mv) { mv = w[1]; am = 1; }
            if (w[2] < mv) { mv = w[2]; am = 2; }
            if (w[3] < mv) { mv = w[3]; am = 3; }
            int nb = topo[tet * 4 + (3 - am)];
            bool move = alive && !inside;
            if (move && nb >= 0) tet = nb;
            alive = alive && (inside || nb >= 0);
        }
        bary4(verts, tetras, p, tet, w);
        bool inside = (w[0] >= -1e-6f) && (w[1] >= -1e-6f) && (w[2] >= -1e-6f) && (w[3] >= -1e-6f);
        bool valid = alive && inside;
        float mk = valid ? 1.0f : 0.0f;

        size_t idx = (size_t)ray * MAXS + k;
        o_ray[idx] = valid ? (float)ray : -1.0f;
        o_tet[idx] = valid ? (float)tet : -1.0f;
        size_t b4 = idx * 4;
        o_bar[b4 + 0] = w[0] * mk;
        o_bar[b4 + 1] = w[1] * mk;
        o_bar[b4 + 2] = w[2] * mk;
        o_bar[b4 + 3] = w[3] * mk;
        size_t b3 = idx * 3;
        o_pos[b3 + 0] = p.x * mk;
        o_pos[b3 + 1] = p.y * mk;
        o_pos[b3 + 2] = p.z * mk;
    }
}

// ---------------------------------------------------------------------------
extern "C" void kernel_launch(void* const* d_in, const int* in_sizes, int n_in,
                              void* d_out, int out_size, void* d_ws, size_t ws_size,
                              hipStream_t stream) {
    const float* verts  = (const float*)d_in[0];   // (1,729,3)
    const float* ro     = (const float*)d_in[1];   // (1,4096,3)
    const float* rd     = (const float*)d_in[2];   // (1,4096,3)
    const int*   tetras = (const int*)d_in[3];     // (3072,4)
    const int*   faces  = (const int*)d_in[4];     // (3072,4,3)
    const int*   topo   = (const int*)d_in[5];     // (3072,4)
    float* out = (float*)d_out;

    float* tminWS = (float*)d_ws;                  // 4096 floats
    int*   fidxWS = (int*)(tminWS + N_RAYS);       // 4096 ints
    float* Bws    = (float*)(fidxWS + N_RAYS);     // 768 tiles * 768 floats = 2.25 MB

    ts_tri_setup<<<(NTRI + 255) / 256, 256, 0, stream>>>(verts, faces, Bws);
    ts_intersect<<<(N_RAYS / 16) * 32 / 256, 256, 0, stream>>>(ro, rd, Bws, tminWS, fidxWS);
    ts_sample<<<(N_RAYS + 255) / 256, 256, 0, stream>>>(verts, ro, rd, tetras, topo,
                                                        tminWS, fidxWS, out);
}